// IALUT_LLVE_34127810134600
// MI455X (gfx1250) — compile-verified
//
#include <hip/hip_runtime.h>

// Problem constants (from reference setup_inputs)
#define VV   33
#define VV2  (VV*VV)          // 1089
#define VV3  (VV*VV*VV)       // 35937
#define VV4  (VV*VV*VV*VV)    // 1185921
#define NPIX (8*512*960)      // T*H*W = 3932160
#define NF   32               // feature dim
#define NR   3                // ranks

typedef __attribute__((ext_vector_type(2))) float v2f;
typedef __attribute__((ext_vector_type(8))) float v8f;

#if defined(__HIP_DEVICE_COMPILE__)
#if __has_builtin(__builtin_amdgcn_wmma_f32_16x16x4_f32)
#define USE_WMMA_LUT 1
#else
#define USE_WMMA_LUT 0
#endif
#else
#define USE_WMMA_LUT 0
#endif

// ---------------------------------------------------------------------------
// Kernel 1: weights[r] = sum_f codes[f] * w_gen[r,f] + b_gen[r]   (3x32 GEMV)
// Written to the tail of d_out (second tuple output of the reference).
// ---------------------------------------------------------------------------
__global__ void weights_kernel(const float* __restrict__ codes,
                               const float* __restrict__ w_gen,
                               const float* __restrict__ b_gen,
                               float* __restrict__ wout) {
    int r = threadIdx.x;
    if (r < NR) {
        float s = b_gen[r];
        #pragma unroll
        for (int f = 0; f < NF; ++f) s += codes[f] * w_gen[r * NF + f];
        wout[r] = s;
    }
}

// ---------------------------------------------------------------------------
// Kernel 2: materialize blended LUT, color-interleaved: lut4[idx*4 + c]
//   lut[c,idx] = w0*W[row,0] + w1*W[row,1] + w2*W[row,2] + b[row],
//   row = c*V^4 + idx.
// One V_WMMA_F32_16X16X4_F32 per 16 entries:
//   A (16x4): row m = [W0, W1, W2, b] of entry e0+m
//     (lanes 0-15 hold K0,K1 in v2f; lanes 16-31 hold K2,K3)
//   B (4x16): every column = [w0, w1, w2, 1]  (same lane pattern)
//   D: all 16 columns identical; lane 0 holds rows 0-7, lane 16 rows 8-15.
// Main loop handles only full groups (no bounds checks); the single tail
// entry (VV4 = 16*74120 + 1) is computed scalar by one lane.
// ---------------------------------------------------------------------------
__global__ void lut_build_kernel(const float* __restrict__ lut_W,
                                 const float* __restrict__ lut_b,
                                 const float* __restrict__ wvec,
                                 float* __restrict__ lut4) {
    const int c       = blockIdx.y;              // color 0..2
    const int lane    = threadIdx.x & 31;
    const int halfsel = lane >> 4;               // 0: lanes 0-15, 1: lanes 16-31
    const int m       = lane & 15;
    const int wpb     = blockDim.x >> 5;
    const int wave    = blockIdx.x * wpb + (threadIdx.x >> 5);
    const int nwaves  = gridDim.x * wpb;

    const float w0 = wvec[0], w1 = wvec[1], w2 = wvec[2];

    // B operand: columns all equal [w0,w1,w2,1]; built once, branchless.
    v2f bmat;
    bmat.x = halfsel ? w2   : w0;
    bmat.y = halfsel ? 1.0f : w1;

    const int nfull = VV4 / 16;                  // 74120 full groups

    for (int grp = wave; grp < nfull; grp += nwaves) {
        const int e0  = grp * 16;
        const int e   = e0 + m;                  // this lane's A row entry
        const int row = c * VV4 + e;

#if USE_WMMA_LUT
        // Branchless per-lane source selection (v_cndmask, no exec juggling):
        //   lanes 0-15 : (W[row,0], W[row,1]);  lanes 16-31: (W[row,2], b[row])
        const float* p0 = lut_W + row * 3 + (halfsel ? 2 : 0);
        const float* p1 = halfsel ? (lut_b + row) : (lut_W + row * 3 + 1);
        v2f amat;
        amat.x = *p0;
        amat.y = *p1;

        v8f acc = {0.f, 0.f, 0.f, 0.f, 0.f, 0.f, 0.f, 0.f};
        acc = __builtin_amdgcn_wmma_f32_16x16x4_f32(
            false, amat, false, bmat, (short)0, acc, false, false);

        // lane 0 -> entries e0..e0+7, lane 16 -> entries e0+8..e0+15 (col 0)
        if (m == 0) {
            float* dst = lut4 + (e0 + halfsel * 8) * 4 + c;
            #pragma unroll
            for (int i = 0; i < 8; ++i) dst[i * 4] = acc[i];
        }
#else
        // VALU fallback (also what the host pass parses): lanes 0-15 each
        // compute their entry fully; lanes 16-31 idle.
        if (halfsel == 0) {
            float v = w0 * lut_W[row * 3 + 0] + w1 * lut_W[row * 3 + 1] +
                      w2 * lut_W[row * 3 + 2] + lut_b[row];
            lut4[e * 4 + c] = v;
        }
#endif
    }

    // Tail entries (exactly 1 for V=33): scalar path, one wave only.
    if (wave == 0) {
        for (int e = nfull * 16 + lane; e < VV4; e += 32) {
            const int row = c * VV4 + e;
            lut4[e * 4 + c] = w0 * lut_W[row * 3 + 0] + w1 * lut_W[row * 3 + 1] +
                              w2 * lut_W[row * 3 + 2] + lut_b[row];
        }
    }
}

// ---------------------------------------------------------------------------
// Kernel 3: quadrilinear 4D-LUT transform. 4 pixels per thread (float4 I/O).
// 16 float4 L2 gathers per pixel (all 3 colors at once); lerp along the
// contiguous innermost LUT dim (channel 3 = intensity, stride 1).
// ---------------------------------------------------------------------------
__global__ void __launch_bounds__(256)
ialut_apply_kernel(const float* __restrict__ videos,
                   const float* __restrict__ inten,
                   const float* __restrict__ lut4,
                   float* __restrict__ out) {
    const int t  = blockIdx.x * blockDim.x + threadIdx.x;
    const int p0 = t * 4;
    if (p0 >= NPIX) return;

    const float4 crv = *(const float4*)(videos + 0 * NPIX + p0);
    const float4 cgv = *(const float4*)(videos + 1 * NPIX + p0);
    const float4 cbv = *(const float4*)(videos + 2 * NPIX + p0);
    const float4 cav = *(const float4*)(inten + p0);

    const float rr[4] = {crv.x, crv.y, crv.z, crv.w};
    const float gg[4] = {cgv.x, cgv.y, cgv.z, cgv.w};
    const float bb[4] = {cbv.x, cbv.y, cbv.z, cbv.w};
    const float aa[4] = {cav.x, cav.y, cav.z, cav.w};

    float oR[4], oG[4], oB[4];

    #pragma unroll
    for (int j = 0; j < 4; ++j) {
        const float ctx[4] = {rr[j], gg[j], bb[j], aa[j]};
        int   i0[4];
        float fr[4];
        #pragma unroll
        for (int k = 0; k < 4; ++k) {
            float x  = fminf(fmaxf(ctx[k], 0.0f), 1.0f) * (float)(VV - 1);
            int   i  = (int)floorf(x);
            i        = i > (VV - 2) ? (VV - 2) : i;   // x>=0 so i>=0 already
            i0[k]    = i;
            fr[k]    = x - (float)i;
        }
        const float c0[2] = {1.0f - fr[0], fr[0]};
        const float c1[2] = {1.0f - fr[1], fr[1]};
        const float c2[2] = {1.0f - fr[2], fr[2]};
        const float f3    = fr[3];

        const int base = i0[0] * VV3 + i0[1] * VV2 + i0[2] * VV + i0[3];

        float accR = 0.0f, accG = 0.0f, accB = 0.0f;
        #pragma unroll
        for (int b0 = 0; b0 < 2; ++b0)
        #pragma unroll
        for (int b1 = 0; b1 < 2; ++b1)
        #pragma unroll
        for (int b2 = 0; b2 < 2; ++b2) {
            const int idx = base + b0 * VV3 + b1 * VV2 + b2 * VV;
            const float w = c0[b0] * c1[b1] * c2[b2];
            const float4 v0 = *(const float4*)(lut4 + idx * 4);
            const float4 v1 = *(const float4*)(lut4 + idx * 4 + 4);
            accR += w * (v0.x + f3 * (v1.x - v0.x));
            accG += w * (v0.y + f3 * (v1.y - v0.y));
            accB += w * (v0.z + f3 * (v1.z - v0.z));
        }
        oR[j] = accR; oG[j] = accG; oB[j] = accB;
    }

    *(float4*)(out + 0 * NPIX + p0) = make_float4(oR[0], oR[1], oR[2], oR[3]);
    *(float4*)(out + 1 * NPIX + p0) = make_float4(oG[0], oG[1], oG[2], oG[3]);
    *(float4*)(out + 2 * NPIX + p0) = make_float4(oB[0], oB[1], oB[2], oB[3]);
}

// ---------------------------------------------------------------------------
// Fallback: fused transform evaluating the rank-3 blend per gather (used only
// if the workspace is too small to hold the materialized LUT).
// ---------------------------------------------------------------------------
__global__ void ialut_apply_fused_kernel(const float* __restrict__ videos,
                                         const float* __restrict__ inten,
                                         const float* __restrict__ lut_W,
                                         const float* __restrict__ lut_b,
                                         const float* __restrict__ wvec,
                                         float* __restrict__ out) {
    const int p = blockIdx.x * blockDim.x + threadIdx.x;
    if (p >= NPIX) return;
    const float w0 = wvec[0], w1 = wvec[1], w2 = wvec[2];

    const float ctx[4] = {videos[p], videos[NPIX + p], videos[2 * NPIX + p], inten[p]};
    int i0[4]; float fr[4];
    #pragma unroll
    for (int k = 0; k < 4; ++k) {
        float x = fminf(fmaxf(ctx[k], 0.0f), 1.0f) * (float)(VV - 1);
        int   i = (int)floorf(x);
        i       = i > (VV - 2) ? (VV - 2) : i;
        i0[k]   = i;
        fr[k]   = x - (float)i;
    }
    const float c0[2] = {1.0f - fr[0], fr[0]};
    const float c1[2] = {1.0f - fr[1], fr[1]};
    const float c2[2] = {1.0f - fr[2], fr[2]};
    const float f3    = fr[3];
    const int base = i0[0] * VV3 + i0[1] * VV2 + i0[2] * VV + i0[3];

    float acc[3] = {0.0f, 0.0f, 0.0f};
    #pragma unroll
    for (int b0 = 0; b0 < 2; ++b0)
    #pragma unroll
    for (int b1 = 0; b1 < 2; ++b1)
    #pragma unroll
    for (int b2 = 0; b2 < 2; ++b2) {
        const int idx = base + b0 * VV3 + b1 * VV2 + b2 * VV;
        const float w = c0[b0] * c1[b1] * c2[b2];
        #pragma unroll
        for (int c = 0; c < 3; ++c) {
            const int r0 = c * VV4 + idx;
            float v0 = w0 * lut_W[r0 * 3] + w1 * lut_W[r0 * 3 + 1] +
                       w2 * lut_W[r0 * 3 + 2] + lut_b[r0];
            float v1 = w0 * lut_W[(r0 + 1) * 3] + w1 * lut_W[(r0 + 1) * 3 + 1] +
                       w2 * lut_W[(r0 + 1) * 3 + 2] + lut_b[r0 + 1];
            acc[c] += w * (v0 + f3 * (v1 - v0));
        }
    }
    out[0 * NPIX + p] = acc[0];
    out[1 * NPIX + p] = acc[1];
    out[2 * NPIX + p] = acc[2];
}

// ---------------------------------------------------------------------------
extern "C" void kernel_launch(void* const* d_in, const int* in_sizes, int n_in,
                              void* d_out, int out_size, void* d_ws, size_t ws_size,
                              hipStream_t stream) {
    const float* videos = (const float*)d_in[0];
    const float* inten  = (const float*)d_in[1];
    const float* codes  = (const float*)d_in[2];
    const float* w_gen  = (const float*)d_in[3];
    const float* b_gen  = (const float*)d_in[4];
    const float* lut_W  = (const float*)d_in[5];
    const float* lut_b  = (const float*)d_in[6];

    float* out  = (float*)d_out;
    float* wvec = out + (size_t)3 * NPIX;   // weights = second tuple output

    // 1) rank weights
    weights_kernel<<<1, 32, 0, stream>>>(codes, w_gen, b_gen, wvec);

    const size_t need = (size_t)VV4 * 4 * sizeof(float);   // ~19 MB
    if (ws_size >= need) {
        float* lut4 = (float*)d_ws;
        // 2) blended LUT via WMMA (rank-3 GEMV, K=4 with bias column)
        dim3 grid(512, 3, 1);
        lut_build_kernel<<<grid, 256, 0, stream>>>(lut_W, lut_b, wvec, lut4);
        // 3) quadrilinear transform, 4 pixels/thread (NPIX/4 = 983040)
        ialut_apply_kernel<<<(NPIX / 4) / 256, 256, 0, stream>>>(videos, inten, lut4, out);
    } else {
        ialut_apply_fused_kernel<<<(NPIX + 255) / 256, 256, 0, stream>>>(
            videos, inten, lut_W, lut_b, wvec, out);
    }
}